// ConvLSTMModel_58437325029778
// MI455X (gfx1250) — compile-verified
//
#include <hip/hip_runtime.h>
#include <math.h>

// ---- problem constants (match reference) ----
#define BB   8
#define TT   10
#define HH   256
#define WW   256
#define CIN  8
#define NG   4          // gates
#define KTOT 84         // 72 x-taps + 9 h-taps + 1 bias + 2 pad
#define WFSTRIDE 88     // padded K stride for fused weights (even, 8B-aligned pairs)

typedef __attribute__((ext_vector_type(2))) float v2f;
typedef __attribute__((ext_vector_type(8))) float v8f;

// Pack fused weight matrix Wf[16][88], gate-major, K contiguous:
//   kk <  72 : Wx[dy][dx][c][g],  kk = (dy*3+dx)*8 + c
//   72..80   : Wh[dy][dx][0][g],  kk-72 = dy*3+dx
//   kk == 81 : bias[g]   (A supplies 1.0 in this K row)
//   else 0 ; gates >= 4 are zero columns
__global__ void convlstm_prep(const float* __restrict__ Wx,
                              const float* __restrict__ Wh,
                              const float* __restrict__ bias,
                              float* __restrict__ Wf) {
    int i = blockIdx.x * blockDim.x + threadIdx.x;
    if (i >= 16 * WFSTRIDE) return;
    int n = i / WFSTRIDE, kk = i % WFSTRIDE;
    float v = 0.0f;
    if (n < NG) {
        if (kk < 72)        v = Wx[kk * NG + n];
        else if (kk < 81)   v = Wh[(kk - 72) * NG + n];
        else if (kk == 81)  v = bias[n];
    }
    Wf[i] = v;
}

// One ConvLSTM timestep. Block = 256 threads (8 waves) = one 1x128 pixel strip.
// Each wave computes z[16 px, 4 gates] via 21 chained v_wmma_f32_16x16x4_f32.
__global__ __launch_bounds__(256)
void convlstm_step(const float* __restrict__ x,      // [B,T,H,W,CIN]
                   const float* __restrict__ Wf,     // [16][88] fused weights
                   const float* __restrict__ hprev,  // [B,H,W]
                   const float* __restrict__ cprev,  // [B,H,W]
                   float* __restrict__ hout,
                   float* __restrict__ cout,
                   int t, int first) {
    __shared__ float xtile[3][130][CIN];   // rows y-1..y+1, cols xbase-1..xbase+128
    __shared__ float htile[3][132];
    __shared__ float wf[16][WFSTRIDE];
    __shared__ float zbuf[8][16][NG];

    const int tid   = threadIdx.x;
    const int blk   = blockIdx.x;              // B*H*2 = 4096 blocks
    const int xb    = blk & 1;
    const int y     = (blk >> 1) & (HH - 1);
    const int b     = blk >> 9;
    const int xbase = xb << 7;                 // 0 or 128

    // ---- stage fused weights ----
    for (int i = tid; i < 16 * WFSTRIDE; i += 256)
        wf[i / WFSTRIDE][i % WFSTRIDE] = Wf[i];

    // ---- stage x halo tile (zero-padded borders) ----
    const float* xt = x + ((size_t)(b * TT + t) * HH * WW * CIN);
    for (int i = tid; i < 3 * 130; i += 256) {
        int r = i / 130, cc = i % 130;
        int gy = y + r - 1, gx = xbase + cc - 1;
        float4 v0 = make_float4(0.f, 0.f, 0.f, 0.f);
        float4 v1 = v0;
        if (gy >= 0 && gy < HH && gx >= 0 && gx < WW) {
            const float4* p = (const float4*)(xt + ((size_t)gy * WW + gx) * CIN);
            v0 = p[0]; v1 = p[1];
        }
        *(float4*)&xtile[r][cc][0] = v0;
        *(float4*)&xtile[r][cc][4] = v1;
    }
    // ---- stage h halo tile ----
    for (int i = tid; i < 3 * 130; i += 256) {
        int r = i / 130, cc = i % 130;
        int gy = y + r - 1, gx = xbase + cc - 1;
        float hv = 0.0f;
        if (!first && gy >= 0 && gy < HH && gx >= 0 && gx < WW)
            hv = hprev[(size_t)(b * HH + gy) * WW + gx];
        htile[r][cc] = hv;
    }
    __syncthreads();

    const int wave = tid >> 5;
    const int lane = tid & 31;
    const int hh   = lane >> 4;          // half-wave selector
    const int lm   = lane & 15;
    const int colbase = wave * 16 + lm;  // pixel column index inside tile (-1 halo origin)

    v8f acc = {};

    // ---- x-conv: 18 WMMA steps, K = 4j..4j+3 all inside the 72 x-taps ----
    // tap = j>>1 is compile-time uniform; channel pair = (j&1)*4 + 2*hh (even).
    #pragma unroll
    for (int j = 0; j < 18; ++j) {
        const int tap = j >> 1;
        const int dy  = tap / 3;
        const int dx  = tap % 3;
        const int cb  = (j & 1) * 4;
        v2f a  = *(const v2f*)&xtile[dy][colbase + dx][cb + 2 * hh];
        v2f bv = *(const v2f*)&wf[lm][4 * j + 2 * hh];
        acc = __builtin_amdgcn_wmma_f32_16x16x4_f32(false, a, false, bv,
                                                    (short)0, acc, false, false);
    }

    // ---- h-conv taps 0..7 : 2 WMMA steps (K = 72..79) ----
    #pragma unroll
    for (int j = 0; j < 2; ++j) {
        int t0 = 4 * j + 2 * hh;                 // 0,2,4,6
        float a0 = htile[t0 / 3][colbase + t0 % 3];
        float a1 = htile[(t0 + 1) / 3][colbase + (t0 + 1) % 3];
        v2f a  = { a0, a1 };
        v2f bv = *(const v2f*)&wf[lm][72 + t0];
        acc = __builtin_amdgcn_wmma_f32_16x16x4_f32(false, a, false, bv,
                                                    (short)0, acc, false, false);
    }

    // ---- final step: K=80 (h tap 8) + K=81 (bias, A=1.0) + K=82,83 zero ----
    {
        float a0 = hh ? 0.0f : htile[2][colbase + 2];
        float a1 = hh ? 0.0f : 1.0f;
        v2f a  = { a0, a1 };
        v2f bv = *(const v2f*)&wf[lm][80 + 2 * hh];
        acc = __builtin_amdgcn_wmma_f32_16x16x4_f32(false, a, false, bv,
                                                    (short)0, acc, false, false);
    }

    // ---- scatter z (D layout: lane=N gate, VGPR r = pixel r / r+8) ----
    if (lm < NG) {
        #pragma unroll
        for (int r = 0; r < 8; ++r)
            zbuf[wave][hh * 8 + r][lm] = acc[r];
    }
    __syncthreads();

    // ---- pointwise gate math, fp32, one lane per pixel ----
    if (lane < 16) {
        int px = xbase + wave * 16 + lane;
        size_t idx = (size_t)(b * HH + y) * WW + px;
        float zi = zbuf[wave][lane][0];
        float zf = zbuf[wave][lane][1];
        float zc = zbuf[wave][lane][2];
        float zo = zbuf[wave][lane][3];
        float ig = fminf(fmaxf(0.2f * zi + 0.5f, 0.0f), 1.0f);
        float fg = fminf(fmaxf(0.2f * zf + 0.5f, 0.0f), 1.0f);
        float og = fminf(fmaxf(0.2f * zo + 0.5f, 0.0f), 1.0f);
        float g  = 1.0f / (1.0f + expf(-zc));
        float c0 = first ? 0.0f : cprev[idx];
        float cn = fg * c0 + ig * g;
        float hn = og * (1.0f / (1.0f + expf(-cn)));
        cout[idx] = cn;
        hout[idx] = hn;
    }
}

extern "C" void kernel_launch(void* const* d_in, const int* in_sizes, int n_in,
                              void* d_out, int out_size, void* d_ws, size_t ws_size,
                              hipStream_t stream) {
    const float* x    = (const float*)d_in[0];
    const float* Wx   = (const float*)d_in[1];
    const float* Wh   = (const float*)d_in[2];
    const float* bias = (const float*)d_in[3];
    float* out = (float*)d_out;

    char* ws = (char*)d_ws;
    const size_t STATE = (size_t)BB * HH * WW;        // 524288 floats
    float* Wf = (float*)ws;                           // 16*88 floats
    float* hA = (float*)(ws + 8192);
    float* cA = hA + STATE;
    float* hB = cA + STATE;
    float* cB = hB + STATE;

    convlstm_prep<<<(16 * WFSTRIDE + 127) / 128, 128, 0, stream>>>(Wx, Wh, bias, Wf);

    const int nblocks = BB * HH * (WW / 128);         // 4096
    for (int t = 0; t < TT; ++t) {
        int even = (t & 1) == 0;
        const float* hp = even ? hB : hA;             // prev (unused when t==0)
        const float* cp = even ? cB : cA;
        float* ho = even ? hA : hB;
        float* co = even ? cA : cB;
        if (t == TT - 1) ho = out;                    // last step: h -> d_out
        convlstm_step<<<nblocks, 256, 0, stream>>>(x, Wf, hp, cp, ho, co,
                                                   t, t == 0 ? 1 : 0);
    }
    (void)in_sizes; (void)n_in; (void)out_size; (void)ws_size;
}